// CIFAR_10_Dis3Uni2_15728170238651
// MI455X (gfx1250) — compile-verified
//
#include <hip/hip_runtime.h>
#include <math.h>

// ---------------------------------------------------------------------------
// CDNA5 (gfx1250) implementation of the CIFAR Dis3Uni2 network.
// Matrix work (densified 2x2 sparse convs + head GEMM) runs on
// V_WMMA_F32_16X16X4_F32. Conv weight tiles are staged into LDS with the
// Tensor Data Mover (tensor_load_to_lds + s_wait_tensorcnt).
// Gather/distance/norm ops are VALU kernels.
// ---------------------------------------------------------------------------

typedef __attribute__((ext_vector_type(2))) float v2f;
typedef __attribute__((ext_vector_type(8))) float v8f;
typedef __attribute__((ext_vector_type(4))) unsigned int v4u;
typedef __attribute__((ext_vector_type(8))) int v8i;
typedef __attribute__((ext_vector_type(4))) int v4i;

#define LN_EPS 1e-5f
#define NEG_INF (-3.402823466e38f)
#define CONV_KC 128   // K-chunk staged in LDS (floats per weight row)

__device__ __forceinline__ v8f wmma_f32_k4(v2f a, v2f b, v8f c) {
  // D = A(16x4,f32) * B(4x16,f32) + C(16x16,f32)
  return __builtin_amdgcn_wmma_f32_16x16x4_f32(
      /*neg_a=*/false, a, /*neg_b=*/false, b,
      /*c_mod=*/(short)0, c, /*reuse_a=*/false, /*reuse_b=*/false);
}

// ---- Tensor Data Mover: 2-D tile (rows x rowlen f32) global -> LDS ----------
// D# layout per CDNA5 ISA section 8.3/8.4 (group0 128b, group1 256b).
__device__ __forceinline__ void tdm_load_2d(unsigned lds_addr, const void* gaddr,
                                            unsigned rowlen, unsigned rows,
                                            unsigned stride_elts) {
  unsigned long long ga = (unsigned long long)gaddr;
  v4u g0;
  g0[0] = 1u;                                   // count=1, user descriptor
  g0[1] = lds_addr;                             // lds_addr [63:32]
  g0[2] = (unsigned)(ga & 0xffffffffu);         // global_addr[31:0]
  g0[3] = (unsigned)((ga >> 32) & 0x01ffffffu)  // global_addr[56:32]
          | (2u << 30);                         // type=2 ("image")
  v8i g1;
  g1[0] = (int)(2u << 16);                      // data_size=2 (4 bytes)
  g1[1] = (int)((rowlen & 0xffffu) << 16);      // tensor_dim0[15:0]
  g1[2] = (int)((rowlen >> 16) | ((rows & 0xffffu) << 16));  // dim0 hi | dim1 lo
  g1[3] = (int)((rows >> 16) | ((rowlen & 0xffffu) << 16));  // dim1 hi | tile_dim0
  g1[4] = (int)(rows & 0xffffu);                // tile_dim1 = rows, tile_dim2 = 0
  g1[5] = (int)stride_elts;                     // tensor_dim0_stride[31:0]
  g1[6] = 0;                                    // stride hi / dim1_stride lo
  g1[7] = 0;
  v4i z4 = {0, 0, 0, 0};
  v8i z8 = {0, 0, 0, 0, 0, 0, 0, 0};
#if __has_builtin(__builtin_amdgcn_tensor_load_to_lds)
  // 6-arg (clang-23 / therock) form: g0, g1, g2, g3, extra group, cpol
  __builtin_amdgcn_tensor_load_to_lds(g0, g1, z4, z4, z8, 0);
#else
  asm volatile("tensor_load_to_lds %0, %1" :: "s"(g0), "s"(g1) : "memory");
#endif
}

__device__ __forceinline__ void wait_tensor0() {
#if __has_builtin(__builtin_amdgcn_s_wait_tensorcnt)
  __builtin_amdgcn_s_wait_tensorcnt(0);
#else
  asm volatile("s_wait_tensorcnt 0x0" ::: "memory");
#endif
}

// ---------------- elementwise helpers ----------------

__global__ void k_zero(float* __restrict__ p, long long n) {
  long long t = (long long)blockIdx.x * blockDim.x + threadIdx.x;
  if (t < n) p[t] = 0.f;
}

// dst[b, co_off+c, p] = src[b, c, p]
__global__ void k_copy_ch(const float* __restrict__ src, float* __restrict__ dst,
                          int Bb, int Cs, int HW, int co_off, int Ctot) {
  long long t = (long long)blockIdx.x * blockDim.x + threadIdx.x;
  long long total = (long long)Bb * Cs * HW;
  if (t >= total) return;
  int p = (int)(t % HW);
  int c = (int)((t / HW) % Cs);
  int b = (int)(t / ((long long)HW * Cs));
  dst[((size_t)b * Ctot + co_off + c) * HW + p] =
      src[((size_t)b * Cs + c) * HW + p];
}

// ---------------- distance op ----------------
// y[b, co_off+o, p] = max_k |x[b, idx[o,k], p] - w[o,k]| + bias[o]
__global__ void k_dist(const float* __restrict__ x, const float* __restrict__ w,
                       const float* __restrict__ bias, const int* __restrict__ idx,
                       float* __restrict__ y, int Bb, int Cin, int O, int HW,
                       int conn, int co_off, int Ctot) {
  long long t = (long long)blockIdx.x * blockDim.x + threadIdx.x;
  long long total = (long long)Bb * O * HW;
  if (t >= total) return;
  int p = (int)(t % HW);
  int o = (int)((t / HW) % O);
  int b = (int)(t / ((long long)HW * O));
  const float* xb = x + (size_t)b * Cin * HW + p;
  const int* irow = idx + (size_t)o * conn;
  const float* wrow = w + (size_t)o * conn;
  float m = NEG_INF;
  for (int k = 0; k < conn; ++k) {
    float v = fabsf(xb[(size_t)irow[k] * HW] - wrow[k]);
    m = fmaxf(m, v);
  }
  y[((size_t)b * Ctot + co_off + o) * HW + p] = m + bias[o];
}

// ---------------- densify sparse 2x2 conv weights ----------------
__global__ void k_densify(const float* __restrict__ wu, const int* __restrict__ idx,
                          float* __restrict__ wd, int O, int Cin, int conn2) {
  long long t = (long long)blockIdx.x * blockDim.x + threadIdx.x;
  long long total = (long long)O * conn2 * 4;
  if (t >= total) return;
  int j = (int)(t & 3);
  int s = (int)((t >> 2) % conn2);
  int o = (int)(t / ((long long)conn2 * 4));
  int c = idx[(size_t)o * conn2 + s];
  atomicAdd(&wd[((size_t)o * Cin + c) * 4 + j],
            wu[((size_t)o * conn2 + s) * 4 + j]);
}

// ---------------- implicit-GEMM 2x2 conv (pad bottom/right) via WMMA ---------
// GEMM view: M = B*H*W, N = O, K = Cin*4 (k = c*4 + kh*2 + kw).
// Block = 4 waves, one shared N-tile; each wave register-blocks 2 M-tiles.
// The 16xK weight tile is TDM-staged into LDS in CONV_KC chunks.
__global__ void __launch_bounds__(128)
k_uni_conv_wmma(const float* __restrict__ x, const float* __restrict__ wd,
                const float* __restrict__ bias, float* __restrict__ y,
                int Bb, int Cin, int O, int H, int Wimg, int co_off, int Ctot) {
  const int HW = H * Wimg;
  const int M = Bb * HW;     // multiple of 128 for all layers
  const int K = Cin * 4;
  const int mblocks = M >> 7;
  int mb = blockIdx.x % mblocks;
  int nt = blockIdx.x / mblocks;
  int wave = threadIdx.x >> 5, lane = threadIdx.x & 31;
  int l16 = lane & 15, hi = lane >> 4;

  __shared__ float bt[CONV_KC * 16];
  unsigned ldsb = (unsigned)(size_t)(void*)bt;   // flat LDS addr[31:0]

  // A-fragment sources: this lane supplies A[m, 2*hi + j]; kh == hi, kw == j.
  const float* xrow[2];
  bool vh[2], vw[2];
#pragma unroll
  for (int q = 0; q < 2; ++q) {
    int m = (mb << 7) + ((wave * 2 + q) << 4) + l16;
    int p = m % HW, b = m / HW;
    int h = p / Wimg, w = p % Wimg;
    int hh = h + hi;
    vh[q] = hh < H;                  // bottom zero-pad
    vw[q] = vh[q] && (w + 1 < Wimg); // right zero-pad
    xrow[q] = x + (size_t)b * Cin * HW + (size_t)hh * Wimg + w;
  }
  const float* gsrc = wd + (size_t)(nt << 4) * K;  // 16 weight rows of this N-tile

  v8f acc0 = {0.f, 0.f, 0.f, 0.f, 0.f, 0.f, 0.f, 0.f};
  v8f acc1 = {0.f, 0.f, 0.f, 0.f, 0.f, 0.f, 0.f, 0.f};

  for (int k0 = 0; k0 < K; k0 += CONV_KC) {
    int cl = K - k0; if (cl > CONV_KC) cl = CONV_KC;   // packed LDS row stride
    if (wave == 0) {
      tdm_load_2d(ldsb, gsrc + k0, (unsigned)cl, 16u, (unsigned)K);
      wait_tensor0();
    }
    __syncthreads();
    for (int kk = 0; kk < cl; kk += 4) {
      v2f bv = *(const v2f*)(&bt[l16 * cl + kk + (hi << 1)]);  // ds_load_b64
      int c = (k0 + kk) >> 2;
      v2f a0, a1;
      a0.x = vh[0] ? xrow[0][(size_t)c * HW] : 0.f;
      a0.y = vw[0] ? xrow[0][(size_t)c * HW + 1] : 0.f;
      a1.x = vh[1] ? xrow[1][(size_t)c * HW] : 0.f;
      a1.y = vw[1] ? xrow[1][(size_t)c * HW + 1] : 0.f;
      acc0 = wmma_f32_k4(a0, bv, acc0);
      acc1 = wmma_f32_k4(a1, bv, acc1);
    }
    __syncthreads();
  }

  int n = (nt << 4) + l16;
  float bn = bias[n];
#pragma unroll
  for (int q = 0; q < 2; ++q) {
    v8f acc = q ? acc1 : acc0;
#pragma unroll
    for (int v = 0; v < 8; ++v) {
      int mm = (mb << 7) + ((wave * 2 + q) << 4) + v + (hi << 3);
      int pp = mm % HW, bb = mm / HW;
      y[((size_t)bb * Ctot + co_off + n) * HW + pp] = acc[v] + bn;
    }
  }
}

// ---------------- plain GEMM  C = A(MxK) * Bt(NxK)^T + bias  via WMMA --------
__global__ void k_gemm_wmma(const float* __restrict__ A, const float* __restrict__ Bt,
                            const float* __restrict__ bias, float* __restrict__ C,
                            int M, int N, int K) {
  int mtiles = M >> 4, ntiles = N >> 4;
  int tile = blockIdx.x * (blockDim.x >> 5) + (threadIdx.x >> 5);
  if (tile >= mtiles * ntiles) return;
  int lane = threadIdx.x & 31;
  int l16 = lane & 15, hi = lane >> 4;
  int mt = tile % mtiles, nt = tile / mtiles;
  const float* arow = A + (size_t)((mt << 4) + l16) * K + (hi << 1);
  const float* brow = Bt + (size_t)((nt << 4) + l16) * K + (hi << 1);
  v8f acc = {0.f, 0.f, 0.f, 0.f, 0.f, 0.f, 0.f, 0.f};
  for (int k0 = 0; k0 < K; k0 += 4) {
    v2f a = *(const v2f*)(arow + k0);
    v2f b = *(const v2f*)(brow + k0);
    acc = wmma_f32_k4(a, b, acc);
  }
  int n = (nt << 4) + l16;
  float bn = bias[n];
#pragma unroll
  for (int v = 0; v < 8; ++v) {
    int mm = (mt << 4) + v + (hi << 3);
    C[(size_t)mm * N + n] = acc[v] + bn;
  }
}

// ---------------- mean-centering ----------------
__global__ void k_mavg_bhw(float* __restrict__ x, int Bb, int C, int HW) {
  int c = blockIdx.x;
  long long n = (long long)Bb * HW;
  __shared__ float r1[256];
  float s = 0.f;
  for (long long i = threadIdx.x; i < n; i += blockDim.x) {
    int b = (int)(i / HW), p = (int)(i % HW);
    s += x[((size_t)b * C + c) * HW + p];
  }
  r1[threadIdx.x] = s;
  __syncthreads();
  for (int st = 128; st > 0; st >>= 1) {
    if ((int)threadIdx.x < st) r1[threadIdx.x] += r1[threadIdx.x + st];
    __syncthreads();
  }
  float m = r1[0] / (float)n;
  for (long long i = threadIdx.x; i < n; i += blockDim.x) {
    int b = (int)(i / HW), p = (int)(i % HW);
    x[((size_t)b * C + c) * HW + p] -= m;
  }
}

__global__ void k_mavg_b(float* __restrict__ x, int Bb, int C, int HW) {
  long long t = (long long)blockIdx.x * blockDim.x + threadIdx.x;
  long long total = (long long)C * HW;
  if (t >= total) return;
  size_t stride = (size_t)C * HW;
  float s = 0.f;
  for (int b = 0; b < Bb; ++b) s += x[b * stride + t];
  float m = s / (float)Bb;
  for (int b = 0; b < Bb; ++b) x[b * stride + t] -= m;
}

// ---------------- layer-norm statistics ----------------
__global__ void k_stats_bhw(const float* __restrict__ x, float* __restrict__ mean,
                            float* __restrict__ den, int Bb, int C, int HW) {
  int c = blockIdx.x;
  long long n = (long long)Bb * HW;
  __shared__ float r1[256], r2[256];
  float s = 0.f, s2 = 0.f;
  for (long long i = threadIdx.x; i < n; i += blockDim.x) {
    int b = (int)(i / HW), p = (int)(i % HW);
    float v = x[((size_t)b * C + c) * HW + p];
    s += v; s2 += v * v;
  }
  r1[threadIdx.x] = s; r2[threadIdx.x] = s2;
  __syncthreads();
  for (int st = 128; st > 0; st >>= 1) {
    if ((int)threadIdx.x < st) {
      r1[threadIdx.x] += r1[threadIdx.x + st];
      r2[threadIdx.x] += r2[threadIdx.x + st];
    }
    __syncthreads();
  }
  if (threadIdx.x == 0) {
    float m = r1[0] / (float)n;
    float var = r2[0] / (float)n - m * m;
    mean[c] = m;
    den[c] = sqrtf(var) + LN_EPS;
  }
}

__global__ void k_stats_b(const float* __restrict__ x, float* __restrict__ mean,
                          float* __restrict__ den, int Bb, int C, int HW) {
  long long t = (long long)blockIdx.x * blockDim.x + threadIdx.x;
  long long total = (long long)C * HW;
  if (t >= total) return;
  size_t stride = (size_t)C * HW;
  float s = 0.f, s2 = 0.f;
  for (int b = 0; b < Bb; ++b) {
    float v = x[b * stride + t];
    s += v; s2 += v * v;
  }
  float m = s / (float)Bb;
  float var = s2 / (float)Bb - m * m;
  mean[t] = m;
  den[t] = sqrtf(var) + LN_EPS;
}

// ---------------- normalize + 2x2 maxpool (fused) ----------------
__global__ void k_pool_norm(const float* __restrict__ x, const float* __restrict__ mean,
                            const float* __restrict__ den, float* __restrict__ y,
                            int Bb, int C, int H, int Wimg, int perPixel) {
  int H2 = H >> 1, W2 = Wimg >> 1, HW2 = H2 * W2, HW = H * Wimg;
  long long t = (long long)blockIdx.x * blockDim.x + threadIdx.x;
  long long total = (long long)Bb * C * HW2;
  if (t >= total) return;
  int p2 = (int)(t % HW2);
  int c = (int)((t / HW2) % C);
  int b = (int)(t / ((long long)HW2 * C));
  int h2 = p2 / W2, w2 = p2 % W2;
  const float* xb = x + ((size_t)b * C + c) * HW;
  float mx = NEG_INF;
#pragma unroll
  for (int dh = 0; dh < 2; ++dh)
#pragma unroll
    for (int dw = 0; dw < 2; ++dw) {
      int p = (2 * h2 + dh) * Wimg + 2 * w2 + dw;
      int si = perPixel ? (c * HW + p) : c;
      float v = (xb[p] - mean[si]) / den[si];
      mx = fmaxf(mx, v);
    }
  y[((size_t)b * C + c) * HW2 + p2] = mx;
}

// ---------------- head: sln(axis 0) + relu, 80->10, softmax ----------------
__global__ void k_head_final(const float* __restrict__ z1, const float* __restrict__ W2,
                             const float* __restrict__ b2, float* __restrict__ out) {
  __shared__ float zs[32 * 80];
  __shared__ float lg[32 * 10];
  int t = threadIdx.x;  // blockDim == 320
  if (t < 80) {
    float s = 0.f, s2 = 0.f;
    for (int b = 0; b < 32; ++b) { float v = z1[b * 80 + t]; s += v; s2 += v * v; }
    float m = s / 32.f;
    float var = s2 / 32.f - m * m;
    float den = sqrtf(var) + LN_EPS;
    for (int b = 0; b < 32; ++b) {
      float v = (z1[b * 80 + t] - m) / den;
      zs[b * 80 + t] = v > 0.f ? v : 0.f;
    }
  }
  __syncthreads();
  {
    int m = t / 10, j = t % 10;
    float s = b2[j];
    for (int k = 0; k < 80; ++k) s += zs[m * 80 + k] * W2[j * 80 + k];
    lg[t] = s;
  }
  __syncthreads();
  if (t < 32) {
    float mx = NEG_INF;
    for (int j = 0; j < 10; ++j) mx = fmaxf(mx, lg[t * 10 + j]);
    float e[10], se = 0.f;
    for (int j = 0; j < 10; ++j) { e[j] = expf(lg[t * 10 + j] - mx); se += e[j]; }
    for (int j = 0; j < 10; ++j) out[t * 10 + j] = e[j] / se;
  }
}

// ---------------------------------------------------------------------------

extern "C" void kernel_launch(void* const* d_in, const int* in_sizes, int n_in,
                              void* d_out, int out_size, void* d_ws, size_t ws_size,
                              hipStream_t stream) {
  (void)in_sizes; (void)n_in; (void)out_size; (void)ws_size;
  const int B = 32;
  auto F = [&](int i) { return (const float*)d_in[i]; };
  auto I = [&](int i) { return (const int*)d_in[i]; };
  auto cdiv = [](long long a, long long b) { return (unsigned)((a + b - 1) / b); };

  // workspace regions (floats)
  float* W  = (float*)d_ws;
  float* R0 = W;                    // 16,777,216  (B1 -> C2 -> C3)
  float* R1 = R0 + 16777216;       //  8,388,608  (Y1a -> Y2a -> Y3a -> BF3 -> B4)
  float* R2 = R1 + 8388608;        //  4,194,304  (BF1 -> BF2 -> Y4 -> BF4)
  float* Wd = R2 + 4194304;        //  8,388,608  densified conv weights
  float* ME = Wd + 8388608;        //    262,144  mean
  float* DE = ME + 262144;         //    262,144  sqrt(var)+eps
  float* Z1 = DE + 262144;         //      2,560  head hidden

  auto run_disuni = [&](const float* in, int Cin, int H, int Wimg,
                        const float* w_dis, const float* b_dis,
                        const float* w_uni, const float* b_uni,
                        const int* idx_dis, const int* idx_uni,
                        int od, int ou, float* out, int co0, int Ctot) {
    int HW = H * Wimg;
    long long td = (long long)B * od * HW;
    k_dist<<<cdiv(td, 256), 256, 0, stream>>>(in, w_dis, b_dis, idx_dis, out,
                                              B, Cin, od, HW, 16, co0, Ctot);
    long long wdn = (long long)ou * Cin * 4;
    k_zero<<<cdiv(wdn, 256), 256, 0, stream>>>(Wd, wdn);
    long long dn = (long long)ou * 32 * 4;
    k_densify<<<cdiv(dn, 256), 256, 0, stream>>>(w_uni, idx_uni, Wd, ou, Cin, 32);
    unsigned blocks = (unsigned)((B * HW / 128) * (ou / 16));
    k_uni_conv_wmma<<<blocks, 128, 0, stream>>>(in, Wd, b_uni, out,
                                                B, Cin, ou, H, Wimg,
                                                co0 + od, Ctot);
  };

  // ---- Stage 1 (32x32) ----
  run_disuni(F(0), 3, 32, 32, F(1), F(2), F(3), F(4), I(33), I(34),
             128, 128, R1, 0, 256);                                  // Y1a
  k_mavg_bhw<<<256, 256, 0, stream>>>(R1, B, 256, 1024);
  run_disuni(R1, 256, 32, 32, F(5), F(6), F(7), F(8), I(35), I(36),
             256, 256, R0, 0, 512);                                  // B1
  k_stats_bhw<<<512, 256, 0, stream>>>(R0, ME, DE, B, 512, 1024);
  k_pool_norm<<<cdiv((long long)B * 512 * 256, 256), 256, 0, stream>>>(
      R0, ME, DE, R2, B, 512, 32, 32, 0);                            // BF1 (16x16)

  // ---- Stage 2 (16x16) ----
  run_disuni(R2, 512, 16, 16, F(9), F(10), F(11), F(12), I(37), I(38),
             512, 512, R1, 0, 1024);                                 // Y2a
  k_mavg_bhw<<<1024, 256, 0, stream>>>(R1, B, 1024, 256);
  run_disuni(R1, 1024, 16, 16, F(13), F(14), F(15), F(16), I(39), I(40),
             512, 512, R0, 512, 1536);                               // B2 -> C2[512:]
  k_copy_ch<<<cdiv((long long)B * 512 * 256, 256), 256, 0, stream>>>(
      R2, R0, B, 512, 256, 0, 1536);                                 // BF1 -> C2[:512]
  k_stats_bhw<<<1536, 256, 0, stream>>>(R0, ME, DE, B, 1536, 256);
  k_pool_norm<<<cdiv((long long)B * 1536 * 64, 256), 256, 0, stream>>>(
      R0, ME, DE, R2, B, 1536, 16, 16, 0);                           // BF2 (8x8)

  // ---- Stage 3 (8x8, norms over batch axis) ----
  run_disuni(R2, 1536, 8, 8, F(17), F(18), F(19), F(20), I(41), I(42),
             1024, 1024, R1, 0, 2048);                               // Y3a
  k_mavg_b<<<cdiv((long long)2048 * 64, 256), 256, 0, stream>>>(R1, B, 2048, 64);
  run_disuni(R1, 2048, 8, 8, F(21), F(22), F(23), F(24), I(43), I(44),
             1024, 1024, R0, 1536, 3584);                            // B3 -> C3[1536:]
  k_copy_ch<<<cdiv((long long)B * 1536 * 64, 256), 256, 0, stream>>>(
      R2, R0, B, 1536, 64, 0, 3584);                                 // BF2 -> C3[:1536]
  k_stats_b<<<cdiv((long long)3584 * 64, 256), 256, 0, stream>>>(R0, ME, DE, B, 3584, 64);
  k_pool_norm<<<cdiv((long long)B * 3584 * 16, 256), 256, 0, stream>>>(
      R0, ME, DE, R1, B, 3584, 8, 8, 1);                             // BF3 (4x4)

  // ---- Stage 4 (4x4, pure distance layers, conn=32) ----
  k_dist<<<cdiv((long long)B * 2048 * 16, 256), 256, 0, stream>>>(
      R1, F(25), F(26), I(45), R2, B, 3584, 2048, 16, 32, 0, 2048);  // Y4
  k_mavg_b<<<cdiv((long long)2048 * 16, 256), 256, 0, stream>>>(R2, B, 2048, 16);
  k_dist<<<cdiv((long long)B * 1024 * 16, 256), 256, 0, stream>>>(
      R2, F(27), F(28), I(46), R1, B, 2048, 1024, 16, 32, 0, 1024);  // B4
  k_stats_b<<<cdiv((long long)1024 * 16, 256), 256, 0, stream>>>(R1, ME, DE, B, 1024, 16);
  k_pool_norm<<<cdiv((long long)B * 1024 * 4, 256), 256, 0, stream>>>(
      R1, ME, DE, R2, B, 1024, 4, 4, 1);                             // BF4 (2x2) = flat 32x4096

  // ---- Head ----
  k_gemm_wmma<<<cdiv(10, 4), 128, 0, stream>>>(R2, F(29), F(30), Z1, 32, 80, 4096);
  k_head_final<<<1, 320, 0, stream>>>(Z1, F(31), F(32), (float*)d_out);
}